// ConcatFlow_47425028883046
// MI455X (gfx1250) — compile-verified
//
#include <hip/hip_runtime.h>

// Problem constants from the reference: B=16, H=540, W=960, fp32 everywhere.
constexpr int Bn    = 16;
constexpr int Hn    = 540;
constexpr int Wn    = 960;
constexpr int PLANE = Hn * Wn;          // 518,400
constexpr int NPIX  = Bn * PLANE;       // 8,294,400
constexpr int NACC4 = (NPIX * 3) / 4;   // 6,220,800 float4 words in accumulator
constexpr int TB    = 256;              // 8 wave32 waves per block
constexpr int XBLK  = (Wn + TB - 1) / TB;   // 4 blocks per image row

// ---------------------------------------------------------------------------
// Kernel 1: zero the accumulator (d_ws) with coalesced b128 stores. Regular
// (RT) stores so the lines are established dirty in L2 — the whole 99.5 MB
// accumulator fits in the 192 MB L2 and the splat atomics then RMW in L2.
// ---------------------------------------------------------------------------
__global__ void da_warp_zero(float4* __restrict__ acc4) {
    int i = blockIdx.x * blockDim.x + threadIdx.x;
    if (i < NACC4) acc4[i] = float4{0.f, 0.f, 0.f, 0.f};
}

// ---------------------------------------------------------------------------
// Kernel 2: bilinear forward splat with hardware fp32 global atomics.
// 3D grid: blockIdx = (x-tile, y, b) -> zero integer divisions.
// Inputs are read-once streams (232 MB total) -> non-temporal loads so they
// do not evict the L2-resident accumulator.
// ---------------------------------------------------------------------------
__global__ void __launch_bounds__(TB)
da_warp_splat(const float* __restrict__ back_flow,  // (B,2,H,W)
              const float* __restrict__ flowBC,     // (B,2,H,W)
              const float* __restrict__ depth,      // (B,1,H,W)
              float* __restrict__ acc)              // (B*H*W, 3) interleaved
{
    const int x = blockIdx.x * TB + threadIdx.x;
    if (x >= Wn) return;
    const int y = blockIdx.y;
    const int b = blockIdx.z;

    const int    rem = y * Wn + x;                    // offset within a plane
    const size_t c0  = (size_t)b * 2 * PLANE + rem;   // channel-0 plane offset

    float bfx = __builtin_nontemporal_load(&back_flow[c0]);
    float bfy = __builtin_nontemporal_load(&back_flow[c0 + PLANE]);
    float fx  = __builtin_nontemporal_load(&flowBC[c0]);
    float fy  = __builtin_nontemporal_load(&flowBC[c0 + PLANE]);
    float d   = __builtin_nontemporal_load(&depth[(size_t)b * PLANE + rem]);

    float dw = __builtin_expf(-d);       // depth weight
    float vx = fx * dw;
    float vy = fy * dw;

    float tx  = (float)x + bfx;
    float ty  = (float)y + bfy;
    float x0f = floorf(tx);
    float y0f = floorf(ty);
    int   x0  = (int)x0f;
    int   y0  = (int)y0f;
    float ax  = tx - x0f;                // in [0,1)
    float ay  = ty - y0f;

    float wxv[2] = {1.f - ax, ax};
    float wyv[2] = {1.f - ay, ay};

#pragma unroll
    for (int oy = 0; oy < 2; ++oy) {
        int yi = y0 + oy;
        if ((unsigned)yi >= (unsigned)Hn) continue;   // invalid tap -> contributes 0
#pragma unroll
        for (int ox = 0; ox < 2; ++ox) {
            int xi = x0 + ox;
            if ((unsigned)xi >= (unsigned)Wn) continue;
            float w = wxv[ox] * wyv[oy];
            size_t o = ((size_t)(b * Hn + yi) * Wn + xi) * 3u;
            // Non-returning agent-scope fp32 atomics -> global_atomic_add_f32
            __hip_atomic_fetch_add(&acc[o + 0], vx * w, __ATOMIC_RELAXED,
                                   __HIP_MEMORY_SCOPE_AGENT);
            __hip_atomic_fetch_add(&acc[o + 1], vy * w, __ATOMIC_RELAXED,
                                   __HIP_MEMORY_SCOPE_AGENT);
            __hip_atomic_fetch_add(&acc[o + 2], dw * w, __ATOMIC_RELAXED,
                                   __HIP_MEMORY_SCOPE_AGENT);
        }
    }
}

// ---------------------------------------------------------------------------
// Kernel 3: normalize accumulator and add flowAB.
// acc reads (b96, contiguous 12B) hit L2; flowAB read + output write are
// non-temporal streams.
// ---------------------------------------------------------------------------
__global__ void __launch_bounds__(TB)
da_warp_finish(const float* __restrict__ acc,     // (B*H*W, 3)
               const float* __restrict__ flowAB,  // (B,2,H,W)
               float* __restrict__ out)           // (B,2,H,W)
{
    const int x = blockIdx.x * TB + threadIdx.x;
    if (x >= Wn) return;
    const int y = blockIdx.y;
    const int b = blockIdx.z;

    const int    rem = y * Wn + x;
    const size_t i   = (size_t)b * PLANE + rem;
    const size_t o3  = i * 3u;

    float nx  = acc[o3 + 0];
    float ny  = acc[o3 + 1];
    float den = acc[o3 + 2];

    // reference: where(den > eps, acc / max(den, eps), 0); den > eps => max = den
    float inv = (den > 1e-6f) ? (1.0f / den) : 0.0f;
    float wx  = nx * inv;
    float wy  = ny * inv;

    const size_t c0 = (size_t)b * 2 * PLANE + rem;
    float a0 = __builtin_nontemporal_load(&flowAB[c0]);
    float a1 = __builtin_nontemporal_load(&flowAB[c0 + PLANE]);
    __builtin_nontemporal_store(wx + a0, &out[c0]);
    __builtin_nontemporal_store(wy + a1, &out[c0 + PLANE]);
}

// ---------------------------------------------------------------------------
// Launcher. Input order (setup_inputs dict): flowAB, back_flowAB, flowBC,
// imgB_depth. Output: (B,2,H,W) fp32. d_ws holds the 3*NPIX float accumulator.
// ---------------------------------------------------------------------------
extern "C" void kernel_launch(void* const* d_in, const int* in_sizes, int n_in,
                              void* d_out, int out_size, void* d_ws, size_t ws_size,
                              hipStream_t stream) {
    const float* flowAB   = (const float*)d_in[0];
    const float* back_fAB = (const float*)d_in[1];
    const float* flowBC   = (const float*)d_in[2];
    const float* depth    = (const float*)d_in[3];
    float*       out      = (float*)d_out;
    float*       acc      = (float*)d_ws;   // needs 3*NPIX*4 = 99.5 MB

    (void)in_sizes; (void)n_in; (void)out_size; (void)ws_size;

    dim3 grid3(XBLK, Hn, Bn);   // (4, 540, 16) blocks of 256 threads

    da_warp_zero<<<(NACC4 + TB - 1) / TB, TB, 0, stream>>>((float4*)acc);
    da_warp_splat<<<grid3, TB, 0, stream>>>(back_fAB, flowBC, depth, acc);
    da_warp_finish<<<grid3, TB, 0, stream>>>(acc, flowAB, out);
}